// TensoRFCPNet_57415122813503
// MI455X (gfx1250) — compile-verified
//
#include <hip/hip_runtime.h>
#include <hip/hip_bf16.h>

#define F_DIM   4
#define R_RANK  100
#define FR      400      // F_DIM * R_RANK
#define N_GRID  500
#define NT_GRID 24

// ---------------------------------------------------------------------------
// Tiled transpose using the CDNA5 async global<->LDS path:
//   global_load_async_to_lds_b32  -> LDS tile (no VGPR staging)
//   s_wait_asynccnt + barrier
//   global_store_async_from_lds_b32 <- transposed LDS tile
// in (rows x cols, row-major) -> out (cols x rows, row-major)
// ---------------------------------------------------------------------------
__global__ __launch_bounds__(256) void transpose_kernel(const float* __restrict__ in,
                                                        float* __restrict__ out,
                                                        int rows, int cols) {
  __shared__ __attribute__((aligned(16))) float tile[32][33];
  const int c0 = blockIdx.x * 32;
  const int r0 = blockIdx.y * 32;
  const int tx = threadIdx.x;
  const int ty = threadIdx.y;
#pragma unroll
  for (int i = ty; i < 32; i += 8) {
    int r = r0 + i, c = c0 + tx;
    if (r < rows && c < cols) {
      unsigned lds = (unsigned)(size_t)(void*)&tile[i][tx];
      unsigned long long ga =
          (unsigned long long)(size_t)(const void*)(in + (size_t)r * cols + c);
      asm volatile("global_load_async_to_lds_b32 %0, %1, off"
                   :: "v"(lds), "v"(ga) : "memory");
    }
  }
  asm volatile("s_wait_asynccnt 0x0" ::: "memory");
  __syncthreads();
#pragma unroll
  for (int i = ty; i < 32; i += 8) {
    int c = c0 + i, r = r0 + tx;
    if (c < cols && r < rows) {
      unsigned lds = (unsigned)(size_t)(void*)&tile[tx][i];
      unsigned long long ga =
          (unsigned long long)(size_t)(void*)(out + (size_t)c * rows + r);
      asm volatile("global_store_async_from_lds_b32 %0, %1, off"
                   :: "v"(ga), "v"(lds) : "memory");
    }
  }
  asm volatile("s_wait_asynccnt 0x0" ::: "memory");
}

// grid_sample(align_corners=False, padding_mode='zeros') tap computation.
__device__ __forceinline__ void tap(float coord, int Nc, int& i0, int& i1,
                                    float& w0, float& w1) {
  float pos = ((coord + 1.0f) * (float)Nc - 1.0f) * 0.5f;
  float f0  = floorf(pos);
  float w   = pos - f0;
  int a = (int)f0;
  int b = a + 1;
  float m0 = (a >= 0 && a < Nc) ? 1.0f : 0.0f;
  float m1 = (b >= 0 && b < Nc) ? 1.0f : 0.0f;
  i0 = a < 0 ? 0 : (a > Nc - 1 ? Nc - 1 : a);
  i1 = b < 0 ? 0 : (b > Nc - 1 ? Nc - 1 : b);
  w0 = (1.0f - w) * m0;
  w1 = w * m1;
}

__device__ __forceinline__ float4 lerp4(float4 a, float4 b, float w0, float w1) {
  float4 r;
  r.x = a.x * w0 + b.x * w1;
  r.y = a.y * w0 + b.y * w1;
  r.z = a.z * w0 + b.z * w1;
  r.w = a.w * w0 + b.w * w1;
  return r;
}

// ---------------------------------------------------------------------------
// Main eval: one wave32 per point. Lane = d*8 + sub: 4 channel groups x 8
// lanes; each lane gathers float4 (b128) spans of the transposed rows,
// r = sub*4 + k*32 (k=0..2, plus sub==0 tail r=96..99). t-grid staged in LDS
// via async b128 copies. 3-step xor reduction within each channel group.
// ---------------------------------------------------------------------------
__global__ __launch_bounds__(256) void tensorf_eval_kernel(
    const float* __restrict__ pts,
    const float* __restrict__ xT, const float* __restrict__ yT,
    const float* __restrict__ zT, const float* __restrict__ tT,
    float* __restrict__ rgb, float* __restrict__ sigma, int npts) {
  __shared__ __attribute__((aligned(16))) float s_t[NT_GRID * FR];  // 38.4 KB

  // ---- async-stage transposed t-grid into LDS (ASYNCcnt path) ----
  {
    unsigned lds_base = (unsigned)(size_t)(void*)&s_t[0];
    unsigned long long gbase = (unsigned long long)(size_t)(const void*)tT;
    const int nchunks = (NT_GRID * FR * 4) / 16;  // 2400 x 16B
    for (int c = threadIdx.x; c < nchunks; c += blockDim.x) {
      unsigned lds = lds_base + (unsigned)c * 16u;
      unsigned long long ga = gbase + (unsigned long long)c * 16ull;
      asm volatile("global_load_async_to_lds_b128 %0, %1, off"
                   :: "v"(lds), "v"(ga) : "memory");
    }
    asm volatile("s_wait_asynccnt 0x0" ::: "memory");
  }
  __syncthreads();

  const int lane  = threadIdx.x & 31;
  const int d     = lane >> 3;   // channel 0..3 (static per lane)
  const int sub   = lane & 7;    // position within channel group
  const int wave  = (int)((blockIdx.x * blockDim.x + threadIdx.x) >> 5);
  const int nwave = (int)((gridDim.x * blockDim.x) >> 5);

  for (int p = wave; p < npts; p += nwave) {
    float4 pt = reinterpret_cast<const float4*>(pts)[p];
    const float x = pt.x, y = pt.y, z = pt.z, t = pt.w;
    const float maskf =
        ((x * x + z * z <= 1.0f) && (y >= -1.0f) && (y <= 1.0f)) ? 1.0f : 0.0f;
    // CROP_R = 1, CY0 = -1, CY1 = 1  =>  xs=x, ys=y, zs=z, ts=2t-1
    const float xs = x;
    const float ys = y;
    const float zs = z;
    const float ts = t * 2.0f - 1.0f;

    int xi0, xi1, yi0, yi1, zi0, zi1, ti0, ti1;
    float wx0, wx1, wy0, wy1, wz0, wz1, wt0, wt1;
    tap(xs, N_GRID, xi0, xi1, wx0, wx1);
    tap(ys, N_GRID, yi0, yi1, wy0, wy1);
    tap(zs, N_GRID, zi0, zi1, wz0, wz1);
    tap(ts, NT_GRID, ti0, ti1, wt0, wt1);

    const float* __restrict__ xr0 = xT + (size_t)xi0 * FR;
    const float* __restrict__ xr1 = xT + (size_t)xi1 * FR;
    const float* __restrict__ yr0 = yT + (size_t)yi0 * FR;
    const float* __restrict__ yr1 = yT + (size_t)yi1 * FR;
    const float* __restrict__ zr0 = zT + (size_t)zi0 * FR;
    const float* __restrict__ zr1 = zT + (size_t)zi1 * FR;
    const float* __restrict__ tr0 = s_t + ti0 * FR;
    const float* __restrict__ tr1 = s_t + ti1 * FR;

    const int fbase = d * R_RANK;
    float4 acc4 = make_float4(0.f, 0.f, 0.f, 0.f);

#pragma unroll
    for (int k = 0; k < 4; ++k) {
      const int r = sub * 4 + k * 32;        // k<3 always valid; k==3 only sub==0
      if (k < 3 || sub == 0) {
        const int f = fbase + r;             // 16B-aligned offsets throughout
        float4 xv = lerp4(*reinterpret_cast<const float4*>(xr0 + f),
                          *reinterpret_cast<const float4*>(xr1 + f), wx0, wx1);
        float4 yv = lerp4(*reinterpret_cast<const float4*>(yr0 + f),
                          *reinterpret_cast<const float4*>(yr1 + f), wy0, wy1);
        float4 zv = lerp4(*reinterpret_cast<const float4*>(zr0 + f),
                          *reinterpret_cast<const float4*>(zr1 + f), wz0, wz1);
        float4 tv = lerp4(*reinterpret_cast<const float4*>(tr0 + f),
                          *reinterpret_cast<const float4*>(tr1 + f), wt0, wt1);
        acc4.x += xv.x * yv.x * zv.x * tv.x;
        acc4.y += xv.y * yv.y * zv.y * tv.y;
        acc4.z += xv.z * yv.z * zv.z * tv.z;
        acc4.w += xv.w * yv.w * zv.w * tv.w;
      }
    }

    float res = (acc4.x + acc4.y) + (acc4.z + acc4.w);
    // reduce across the 8 lanes of this channel group (wave32 butterfly)
    res += __shfl_xor(res, 4, 32);
    res += __shfl_xor(res, 2, 32);
    res += __shfl_xor(res, 1, 32);

    if (sub == 0) {
      if (d == 0) {
        sigma[p] = expf(res) * maskf;
      } else {
        rgb[p * 3 + (d - 1)] = 1.0f / (1.0f + expf(-res));
      }
    }
  }
}

extern "C" void kernel_launch(void* const* d_in, const int* in_sizes, int n_in,
                              void* d_out, int out_size, void* d_ws, size_t ws_size,
                              hipStream_t stream) {
  (void)n_in; (void)out_size; (void)ws_size;
  // setup_inputs order: pts, viewdirs, xvec, yvec, zvec, tvec, iteration
  const float* pts  = (const float*)d_in[0];
  const float* xvec = (const float*)d_in[2];
  const float* yvec = (const float*)d_in[3];
  const float* zvec = (const float*)d_in[4];
  const float* tvec = (const float*)d_in[5];
  const int npts = in_sizes[0] / 4;  // 65536

  float* ws = (float*)d_ws;
  float* xT = ws;                               // 500*400 floats
  float* yT = ws + (size_t)FR * N_GRID;         // +200000
  float* zT = ws + 2 * (size_t)FR * N_GRID;     // +400000
  float* tT = ws + 3 * (size_t)FR * N_GRID;     // +600000 (byte off 2.4e6, 16B aligned)

  float* rgb = (float*)d_out;                   // npts*3 (reference return order)
  float* sig = (float*)d_out + (size_t)npts * 3;

  dim3 tb(32, 8);
  transpose_kernel<<<dim3((N_GRID + 31) / 32, (FR + 31) / 32), tb, 0, stream>>>(xvec, xT, FR, N_GRID);
  transpose_kernel<<<dim3((N_GRID + 31) / 32, (FR + 31) / 32), tb, 0, stream>>>(yvec, yT, FR, N_GRID);
  transpose_kernel<<<dim3((N_GRID + 31) / 32, (FR + 31) / 32), tb, 0, stream>>>(zvec, zT, FR, N_GRID);
  transpose_kernel<<<dim3((NT_GRID + 31) / 32, (FR + 31) / 32), tb, 0, stream>>>(tvec, tT, FR, NT_GRID);

  // 512 blocks x 8 waves = 4096 waves -> 16 points per wave
  tensorf_eval_kernel<<<512, 256, 0, stream>>>(pts, xT, yT, zT, tT, rgb, sig, npts);
}